// SpectralLinear_37606733643881
// MI455X (gfx1250) — compile-verified
//
#include <hip/hip_runtime.h>
#include <math.h>
#include <stdint.h>

typedef __attribute__((ext_vector_type(2))) float v2f;
typedef __attribute__((ext_vector_type(8))) float v8f;

#define NDIM 1024
#define BATCH 16384

// ---------------------------------------------------------------------------
// CDNA5 async global->LDS DMA (ASYNCcnt-tracked, CDNA5 ISA)
// ---------------------------------------------------------------------------
__device__ __forceinline__ void async_load_f4(const float* g, const float* lds) {
    unsigned off = (unsigned)(uintptr_t)lds;   // low 32 bits = LDS offset
    asm volatile("global_load_async_to_lds_b128 %0, %1, off"
                 :: "v"(off), "v"(g) : "memory");
}
__device__ __forceinline__ void wait_async0() {
    asm volatile("s_wait_asynccnt 0x0" ::: "memory");
}

// ---------------------------------------------------------------------------
// Apply triu mask: Um[i][j] = (j >= i) ? U[i][j] : 0
// ---------------------------------------------------------------------------
__global__ void triu_mask_kernel(const float* __restrict__ src, float* __restrict__ dst) {
    int idx = blockIdx.x * 256 + threadIdx.x;
    int i = idx >> 10;
    int j = idx & 1023;
    float v = src[idx];
    dst[idx] = (j >= i) ? v : 0.0f;
}

// ---------------------------------------------------------------------------
// tau[i] = 2 / (u_i . u_i), one wave per row
// ---------------------------------------------------------------------------
__global__ void tau_kernel(const float* __restrict__ Um, float* __restrict__ tau) {
    int wave = (blockIdx.x * 256 + threadIdx.x) >> 5;
    int lane = threadIdx.x & 31;
    if (wave >= NDIM) return;
    const float* row = Um + (size_t)wave * NDIM;
    float acc = 0.0f;
    for (int k = lane; k < NDIM; k += 32) {
        float v = row[k];
        acc += v * v;
    }
    #pragma unroll
    for (int off = 16; off > 0; off >>= 1) acc += __shfl_xor(acc, off, 32);
    if (lane == 0) tau[wave] = 2.0f / acc;
}

// ---------------------------------------------------------------------------
// sigmas = 2r*(sigmoid(p)-0.5) + (min+r), r = 0.45, mean = 0.55
// ---------------------------------------------------------------------------
__global__ void sigma_kernel(const float* __restrict__ p, float* __restrict__ sig) {
    int i = blockIdx.x * 256 + threadIdx.x;
    if (i < NDIM) {
        float s = 1.0f / (1.0f + expf(-p[i]));
        sig[i] = 0.9f * (s - 0.5f) + 0.55f;
    }
}

// ---------------------------------------------------------------------------
// Upper-triangular inverse T = R^{-1}. One wave per column j.
// ---------------------------------------------------------------------------
__global__ void __launch_bounds__(256) trinv_kernel(const float* __restrict__ R,
                                                    float* __restrict__ T) {
    __shared__ float colbuf[8][NDIM];
    int wave = threadIdx.x >> 5;
    int lane = threadIdx.x & 31;
    int j = blockIdx.x * 8 + wave;
    float* col = colbuf[wave];

    for (int i = j + 1 + lane; i < NDIM; i += 32) T[(size_t)i * NDIM + j] = 0.0f;

    float xjj = 1.0f / R[(size_t)j * NDIM + j];
    col[j] = xjj;
    if (lane == 0) T[(size_t)j * NDIM + j] = xjj;
    __builtin_amdgcn_wave_barrier();

    for (int i = j - 1; i >= 0; --i) {
        const float* Ri = R + (size_t)i * NDIM;
        float dot = 0.0f;
        for (int k = i + 1 + lane; k <= j; k += 32) dot += Ri[k] * col[k];
        #pragma unroll
        for (int off = 16; off > 0; off >>= 1) dot += __shfl_xor(dot, off, 32);
        float x = -dot / Ri[i];
        col[i] = x;
        if (lane == 0) T[(size_t)i * NDIM + j] = x;
        __builtin_amdgcn_wave_barrier();
    }
}

// ---------------------------------------------------------------------------
// Generic WMMA f32 GEMM: C = epilogue( op(A) * op(B) ), 128x128x32 tiles,
// 8 wave32/block, double-buffered LDS. If both operands are non-transposed, both are staged
// with GLOBAL_LOAD_ASYNC_TO_LDS_B128 (overlaps HBM DMA with WMMA);
// otherwise both use register-pipelined load/store (transposing TA/TB=1 ones).
//   TA/TB: 0 = normal, 1 = transposed operand
//   EPI: 0 plain | 1: I-acc | 2: aux[r]*(I-acc) | 3: acc+bias[c]
//        4: Gram->R: diag=1/aux[r], upper=acc, lower=0
// ---------------------------------------------------------------------------
template <int TA, int TB, int EPI>
__global__ void __launch_bounds__(256) gemm_kernel(
    const float* __restrict__ A, const float* __restrict__ B, float* __restrict__ C,
    int M, int N, int K, const float* __restrict__ aux, const float* __restrict__ bias)
{
    constexpr bool ASYNC = (TA == 0 && TB == 0);
    __shared__ float As[2][128][36];   // stride 36: 16B rows, conflict-free frags
    __shared__ float Bs[2][32][132];

    const int tid  = threadIdx.x;
    const int lane = tid & 31;
    const int wave = tid >> 5;
    const int wr   = wave & 3;   // 32-row group
    const int wc   = wave >> 2;  // 64-col group
    const int m0   = blockIdx.x * 128;
    const int n0   = blockIdx.y * 128;
    const int half = lane >> 4;
    const int l16  = lane & 15;

    v8f acc[2][4];
    #pragma unroll
    for (int a = 0; a < 2; ++a)
        #pragma unroll
        for (int b = 0; b < 4; ++b)
            #pragma unroll
            for (int q = 0; q < 8; ++q) acc[a][b][q] = 0.0f;

    float4 ra[4], rb[4];

    auto issue_async = [&](int kb, int buf) {
        const int k0 = kb << 5;
        #pragma unroll
        for (int it = 0; it < 4; ++it) {
            int q = tid + it * 256;
            int r = q >> 3, c4 = (q & 7) << 2;
            async_load_f4(A + (size_t)(m0 + r) * K + k0 + c4, &As[buf][r][c4]);
        }
        #pragma unroll
        for (int it = 0; it < 4; ++it) {
            int q = tid + it * 256;
            int kr = q >> 5, n4 = (q & 31) << 2;
            async_load_f4(B + (size_t)(k0 + kr) * N + n0 + n4, &Bs[buf][kr][n4]);
        }
    };

    auto load_regs = [&](int kb) {
        const int k0 = kb << 5;
        #pragma unroll
        for (int it = 0; it < 4; ++it) {
            int q = tid + it * 256;
            if (TA == 0) { int r = q >> 3, c4 = (q & 7) << 2;
                ra[it] = *(const float4*)(A + (size_t)(m0 + r) * K + k0 + c4); }
            else { int kr = q >> 5, m4 = (q & 31) << 2;
                ra[it] = *(const float4*)(A + (size_t)(k0 + kr) * M + m0 + m4); }
            if (TB == 0) { int kr = q >> 5, n4 = (q & 31) << 2;
                rb[it] = *(const float4*)(B + (size_t)(k0 + kr) * N + n0 + n4); }
            else { int nr = q >> 3, c4 = (q & 7) << 2;
                rb[it] = *(const float4*)(B + (size_t)(n0 + nr) * K + k0 + c4); }
        }
    };

    auto store_lds = [&](int buf) {
        #pragma unroll
        for (int it = 0; it < 4; ++it) {
            int q = tid + it * 256;
            if (TA == 0) { int r = q >> 3, c4 = (q & 7) << 2;
                As[buf][r][c4 + 0] = ra[it].x; As[buf][r][c4 + 1] = ra[it].y;
                As[buf][r][c4 + 2] = ra[it].z; As[buf][r][c4 + 3] = ra[it].w; }
            else { int kr = q >> 5, m4 = (q & 31) << 2;
                As[buf][m4 + 0][kr] = ra[it].x; As[buf][m4 + 1][kr] = ra[it].y;
                As[buf][m4 + 2][kr] = ra[it].z; As[buf][m4 + 3][kr] = ra[it].w; }
            if (TB == 0) { int kr = q >> 5, n4 = (q & 31) << 2;
                Bs[buf][kr][n4 + 0] = rb[it].x; Bs[buf][kr][n4 + 1] = rb[it].y;
                Bs[buf][kr][n4 + 2] = rb[it].z; Bs[buf][kr][n4 + 3] = rb[it].w; }
            else { int nr = q >> 3, c4 = (q & 7) << 2;
                Bs[buf][c4 + 0][nr] = rb[it].x; Bs[buf][c4 + 1][nr] = rb[it].y;
                Bs[buf][c4 + 2][nr] = rb[it].z; Bs[buf][c4 + 3][nr] = rb[it].w; }
        }
    };

    auto compute = [&](int buf) {
        #pragma unroll
        for (int k = 0; k < 32; k += 4) {
            v2f afr[2], bfr[4];
            #pragma unroll
            for (int s = 0; s < 2; ++s) {
                int r = wr * 32 + s * 16 + l16;
                afr[s] = *(const v2f*)&As[buf][r][k + half * 2];   // one ds_load_b64
            }
            #pragma unroll
            for (int s = 0; s < 4; ++s) {
                int c = wc * 64 + s * 16 + l16;
                bfr[s][0] = Bs[buf][k + half * 2][c];
                bfr[s][1] = Bs[buf][k + half * 2 + 1][c];
            }
            #pragma unroll
            for (int a = 0; a < 2; ++a)
                #pragma unroll
                for (int b = 0; b < 4; ++b)
                    acc[a][b] = __builtin_amdgcn_wmma_f32_16x16x4_f32(
                        false, afr[a], false, bfr[b], (short)0, acc[a][b], false, false);
        }
    };

    const int KT = K >> 5;
    // prologue: stage tile 0 into buffer 0
    if constexpr (ASYNC) { issue_async(0, 0); wait_async0(); }
    else                 { load_regs(0); store_lds(0); }
    __syncthreads();

    for (int kb = 0; kb < KT; ++kb) {
        const int cur = kb & 1, nxt = cur ^ 1;
        const bool more = (kb + 1 < KT);
        if (more) {
            if constexpr (ASYNC) {
                issue_async(kb + 1, nxt);          // DMA overlaps with WMMA below
            } else {
                load_regs(kb + 1);                 // loads in flight during compute
                if (kb + 2 < KT) {                 // L2 prefetch 2 tiles ahead
                    int kn = (kb + 2) << 5;
                    if (TA == 0) { int r = tid >> 3, c4 = (tid & 7) << 2;
                        __builtin_prefetch(A + (size_t)(m0 + r) * K + kn + c4, 0, 3); }
                    else         { int kr = tid >> 5, m4 = (tid & 31) << 2;
                        __builtin_prefetch(A + (size_t)(kn + kr) * M + m0 + m4, 0, 3); }
                    if (TB == 0) { int kr = tid >> 5, n4 = (tid & 31) << 2;
                        __builtin_prefetch(B + (size_t)(kn + kr) * N + n0 + n4, 0, 3); }
                    else         { int nr = tid >> 3, c4 = (tid & 7) << 2;
                        __builtin_prefetch(B + (size_t)(n0 + nr) * K + kn + c4, 0, 3); }
                }
            }
        }
        compute(cur);
        if (more) {
            if constexpr (ASYNC) wait_async0();    // next tile landed in LDS
            else                 store_lds(nxt);
        }
        __syncthreads();
    }

    // epilogue: C/D layout VGPR v -> M = v + 8*half, lane&15 -> N
    #pragma unroll
    for (int a = 0; a < 2; ++a) {
        #pragma unroll
        for (int b = 0; b < 4; ++b) {
            int colg = n0 + wc * 64 + b * 16 + l16;
            #pragma unroll
            for (int v = 0; v < 8; ++v) {
                int rowg = m0 + wr * 32 + a * 16 + half * 8 + v;
                float val = acc[a][b][v];
                if (EPI == 1)      val = ((rowg == colg) ? 1.0f : 0.0f) - val;
                else if (EPI == 2) val = aux[rowg] * (((rowg == colg) ? 1.0f : 0.0f) - val);
                else if (EPI == 3) val = val + bias[colg];
                else if (EPI == 4) val = (rowg == colg) ? (1.0f / aux[rowg])
                                         : ((rowg < colg) ? val : 0.0f);
                C[(size_t)rowg * N + colg] = val;
            }
        }
    }
}

// ---------------------------------------------------------------------------
// Launch: compact-WY reformulation
//   R_U = diag(1/tau) + strict_upper(Um Um^T);  T_U = R_U^{-1}
//   A = I - Um^T T_U Um;  M = diag(sig) (I - Vm^T T_V^T Vm);  W = A M
//   out = x W + bias
// ---------------------------------------------------------------------------
extern "C" void kernel_launch(void* const* d_in, const int* in_sizes, int n_in,
                              void* d_out, int out_size, void* d_ws, size_t ws_size,
                              hipStream_t stream)
{
    const float* x    = (const float*)d_in[0];
    const float* U    = (const float*)d_in[1];
    const float* V    = (const float*)d_in[2];
    const float* p    = (const float*)d_in[3];
    const float* bias = (const float*)d_in[4];
    float* out = (float*)d_out;
    float* ws  = (float*)d_ws;

    const size_t MM = (size_t)NDIM * NDIM;
    float* Umm  = ws;
    float* Vmm  = Umm + MM;
    float* RU   = Vmm + MM;
    float* RV   = RU + MM;
    float* TU   = RV + MM;
    float* TV   = TU + MM;
    float* P    = TV + MM;
    float* Amat = P + MM;
    float* Mmat = Amat + MM;
    float* W    = Mmat + MM;
    float* tauU = W + MM;
    float* tauV = tauU + NDIM;
    float* sig  = tauV + NDIM;

    triu_mask_kernel<<<(int)(MM / 256), 256, 0, stream>>>(U, Umm);
    triu_mask_kernel<<<(int)(MM / 256), 256, 0, stream>>>(V, Vmm);
    tau_kernel<<<NDIM / 8, 256, 0, stream>>>(Umm, tauU);
    tau_kernel<<<NDIM / 8, 256, 0, stream>>>(Vmm, tauV);
    sigma_kernel<<<NDIM / 256, 256, 0, stream>>>(p, sig);

    dim3 g1(NDIM / 128, NDIM / 128);
    // Gram + R build:  R = diag(1/tau) + strict_upper(Um Um^T)
    gemm_kernel<0, 1, 4><<<g1, 256, 0, stream>>>(Umm, Umm, RU, NDIM, NDIM, NDIM, tauU, nullptr);
    gemm_kernel<0, 1, 4><<<g1, 256, 0, stream>>>(Vmm, Vmm, RV, NDIM, NDIM, NDIM, tauV, nullptr);
    // Triangular inverses
    trinv_kernel<<<NDIM / 8, 256, 0, stream>>>(RU, TU);
    trinv_kernel<<<NDIM / 8, 256, 0, stream>>>(RV, TV);
    // A = I - Um^T (T_U Um)
    gemm_kernel<0, 0, 0><<<g1, 256, 0, stream>>>(TU, Umm, P, NDIM, NDIM, NDIM, nullptr, nullptr);
    gemm_kernel<1, 0, 1><<<g1, 256, 0, stream>>>(Umm, P, Amat, NDIM, NDIM, NDIM, nullptr, nullptr);
    // M = diag(sig) (I - Vm^T (T_V^T Vm))
    gemm_kernel<1, 0, 0><<<g1, 256, 0, stream>>>(TV, Vmm, P, NDIM, NDIM, NDIM, nullptr, nullptr);
    gemm_kernel<1, 0, 2><<<g1, 256, 0, stream>>>(Vmm, P, Mmat, NDIM, NDIM, NDIM, sig, nullptr);
    // W = A M
    gemm_kernel<0, 0, 0><<<g1, 256, 0, stream>>>(Amat, Mmat, W, NDIM, NDIM, NDIM, nullptr, nullptr);
    // out = x W + bias   (16384 x 1024 x 1024, HBM-bound: ~6 us floor)
    dim3 g2(BATCH / 128, NDIM / 128);
    gemm_kernel<0, 0, 3><<<g2, 256, 0, stream>>>(x, W, out, BATCH, NDIM, NDIM, nullptr, bias);
}